// BallUnpooling_72292889526276
// MI455X (gfx1250) — compile-verified
//
#include <hip/hip_runtime.h>

typedef float v2f __attribute__((ext_vector_type(2)));
typedef float v8f __attribute__((ext_vector_type(8)));

#define N_PARENT 65536
#define STRIDE   8
#define IN_DIM   32
#define OUT_DIM  16
#define MV       16
#define S_IN     (IN_DIM + STRIDE)     // 40
#define OC       (STRIDE * OUT_DIM)    // 128
#define EPS      1e-6f

__global__ __launch_bounds__(256) void ball_unpool_wmma(
    const float* __restrict__ mv,            // [N,32,16]
    const float* __restrict__ sc,            // [N,32]
    const float* __restrict__ pos,           // [N,3]
    const float* __restrict__ children_mv,   // [N*8,16,16]
    const float* __restrict__ children_sc,   // [N*8,16]
    const float* __restrict__ children_pos,  // [N*8,3]
    const float* __restrict__ W_mv,          // [128,32]
    const float* __restrict__ W_s2mv,        // [128,40]
    const float* __restrict__ W_s,           // [128,40]
    const float* __restrict__ W_mv2s,        // [128,32]
    float* __restrict__ out_mv,              // [N*8,16,16]
    float* __restrict__ out_sc)              // [N*8,16]
{
    __shared__ float lds_sc [8][S_IN];    // sc_cat per wave
    __shared__ float lds_mv0[8][IN_DIM];  // mv[n, :, 0]
    __shared__ float lds_add0[8][OC];     // W_s2mv @ sc_cat  (into blade 0)
    __shared__ float lds_scp [8][OC];     // scalar-channel pre-norm output

    const int wave = threadIdx.x >> 5;
    const int lane = threadIdx.x & 31;
    const int n    = blockIdx.x * 8 + wave;

    const int g   = lane >> 4;   // lane half
    const int col = lane & 15;   // column / M-row within half

    // ---- stage rotation-invariant scalar features in LDS ----
    lds_sc [wave][lane] = sc[n * IN_DIM + lane];
    lds_mv0[wave][lane] = mv[(size_t)(n * IN_DIM + lane) * MV];
    if (lane < STRIDE) {
        const float* cp = children_pos + (size_t)(n * STRIDE + lane) * 3;
        const float* pp = pos + (size_t)n * 3;
        float dx = cp[0] - pp[0], dy = cp[1] - pp[1], dz = cp[2] - pp[2];
        lds_sc[wave][IN_DIM + lane] = sqrtf(dx * dx + dy * dy + dz * dz);
    }
    __syncthreads();

    // ---- per-lane scalar dot products: outputs o = lane + 32*m ----
    for (int m = 0; m < 4; ++m) {
        int o = lane + 32 * m;
        float a0 = 0.f, sp = 0.f;
        const float4* w2 = (const float4*)(W_s2mv + o * S_IN);   // 160B rows, 16B aligned
        const float4* ws = (const float4*)(W_s    + o * S_IN);
        #pragma unroll
        for (int q = 0; q < S_IN / 4; ++q) {
            float4 a = w2[q], b = ws[q];
            float s0 = lds_sc[wave][4*q+0], s1 = lds_sc[wave][4*q+1];
            float s2 = lds_sc[wave][4*q+2], s3 = lds_sc[wave][4*q+3];
            a0 += s0*a.x + s1*a.y + s2*a.z + s3*a.w;
            sp += s0*b.x + s1*b.y + s2*b.z + s3*b.w;
        }
        const float4* wm = (const float4*)(W_mv2s + o * IN_DIM);
        #pragma unroll
        for (int q = 0; q < IN_DIM / 4; ++q) {
            float4 a = wm[q];
            sp += lds_mv0[wave][4*q+0]*a.x + lds_mv0[wave][4*q+1]*a.y
                + lds_mv0[wave][4*q+2]*a.z + lds_mv0[wave][4*q+3]*a.w;
        }
        lds_add0[wave][o] = a0;
        lds_scp [wave][o] = sp;
    }
    __syncthreads();

    // ---- B fragments: B = mv[n] (32x16), held once, reused by all 8 tiles ----
    // VGPR0: K=4k+2g ; VGPR1: K=4k+2g+1 ; column = lane%16
    v2f bfrag[8];
    {
        const float* mb = mv + (size_t)n * IN_DIM * MV;
        #pragma unroll
        for (int k = 0; k < 8; ++k) {
            int krow = 4 * k + 2 * g;
            bfrag[k][0] = mb[(krow + 0) * MV + col];
            bfrag[k][1] = mb[(krow + 1) * MV + col];
        }
    }

    // ---- per output tile t == child t: WMMA chain, residual, norms ----
    const int rowbase = 8 * g;   // C/D rows held by this lane: rowbase + r
    for (int t = 0; t < STRIDE; ++t) {
        // seed C with scalar->blade0 coupling in column e=0
        v8f c;
        #pragma unroll
        for (int r = 0; r < 8; ++r) {
            float v = lds_add0[wave][16 * t + rowbase + r];
            c[r] = (col == 0) ? v : 0.0f;
        }
        // A row for this lane: W_mv[16t + lane%16, :]
        const float* wa = W_mv + (size_t)(16 * t + col) * IN_DIM;
        #pragma unroll
        for (int k = 0; k < 8; ++k) {
            v2f a = *(const v2f*)(wa + 4 * k + 2 * g);   // consecutive K pair
            c = __builtin_amdgcn_wmma_f32_16x16x4_f32(false, a, false, bfrag[k],
                                                      (short)0, c, false, false);
        }

        // residual + EquiLayerNorm (RMS over 256 elements of the 16x16 tile)
        size_t child = (size_t)n * STRIDE + t;
        const float* cmv = children_mv + child * (OUT_DIM * MV);
        float x[8], ss = 0.f;
        #pragma unroll
        for (int r = 0; r < 8; ++r) {
            x[r] = cmv[(rowbase + r) * MV + col] + c[r];
            ss += x[r] * x[r];
        }
        #pragma unroll
        for (int msk = 16; msk >= 1; msk >>= 1)
            ss += __shfl_xor(ss, msk, 32);
        float inv = 1.0f / sqrtf(ss * (1.0f / 256.0f) + EPS);
        float* omv = out_mv + child * (OUT_DIM * MV);
        #pragma unroll
        for (int r = 0; r < 8; ++r)
            omv[(rowbase + r) * MV + col] = x[r] * inv;

        // scalar channels: affine-free LayerNorm over 16 values (lanes 0..15)
        if (lane < OUT_DIM) {
            float s = children_sc[child * OUT_DIM + lane] + lds_scp[wave][16 * t + lane];
            float mu = s;
            #pragma unroll
            for (int msk = 8; msk >= 1; msk >>= 1)
                mu += __shfl_xor(mu, msk, 32);
            mu *= (1.0f / 16.0f);
            float d = s - mu;
            float v = d * d;
            #pragma unroll
            for (int msk = 8; msk >= 1; msk >>= 1)
                v += __shfl_xor(v, msk, 32);
            v *= (1.0f / 16.0f);
            out_sc[child * OUT_DIM + lane] = d / sqrtf(v + EPS);
        }
    }
}

extern "C" void kernel_launch(void* const* d_in, const int* in_sizes, int n_in,
                              void* d_out, int out_size, void* d_ws, size_t ws_size,
                              hipStream_t stream) {
    const float* mv           = (const float*)d_in[0];
    const float* sc           = (const float*)d_in[1];
    const float* pos          = (const float*)d_in[2];
    const float* children_mv  = (const float*)d_in[3];
    const float* children_sc  = (const float*)d_in[4];
    const float* children_pos = (const float*)d_in[5];
    const float* W_mv         = (const float*)d_in[6];
    const float* W_s2mv       = (const float*)d_in[7];
    const float* W_s          = (const float*)d_in[8];
    const float* W_mv2s       = (const float*)d_in[9];

    float* out_mv = (float*)d_out;
    float* out_sc = (float*)d_out + (size_t)N_PARENT * STRIDE * OUT_DIM * MV;

    dim3 grid(N_PARENT / 8);   // 8 waves (parents) per 256-thread block
    dim3 block(256);
    ball_unpool_wmma<<<grid, block, 0, stream>>>(
        mv, sc, pos, children_mv, children_sc, children_pos,
        W_mv, W_s2mv, W_s, W_mv2s, out_mv, out_sc);
}